// ImplicitQuantilePooling1d_67774583931261
// MI455X (gfx1250) — compile-verified
//
#include <hip/hip_runtime.h>

// Problem constants (from reference)
#define B_     16
#define C_     128
#define L_     16384
#define K_     8
#define S_     4
#define W_     4095        // (L-K)/S + 1
#define ITERS_ 20
#define TPB    256         // 8 wave32 per block
#define CHUNKS 16          // 16*256 = 4096 window slots >= W_
// last valid 16B load start within a row, in bytes: (L-4)*4
#define ROW_CLAMP 65520u

// Eight gfx1250 v_tanh_f32 (TRANS) ops in one block. Each tanh is independent,
// so consecutive tanhs satisfy the TRANS RAW-hazard spacing for one another
// (ISA 7.4: 1 independent op or V_NOP before a trans result is consumed);
// a single trailing v_nop covers the last one against external consumers.
// "+v" makes each op in-place (dst == src), so no output/input aliasing issues.
__device__ __forceinline__ void tanh8(float& a0, float& a1, float& a2, float& a3,
                                      float& a4, float& a5, float& a6, float& a7) {
    asm("v_tanh_f32 %0, %0\n\t"
        "v_tanh_f32 %1, %1\n\t"
        "v_tanh_f32 %2, %2\n\t"
        "v_tanh_f32 %3, %3\n\t"
        "v_tanh_f32 %4, %4\n\t"
        "v_tanh_f32 %5, %5\n\t"
        "v_tanh_f32 %6, %6\n\t"
        "v_tanh_f32 %7, %7\n\t"
        "v_nop"
        : "+v"(a0), "+v"(a1), "+v"(a2), "+v"(a3),
          "+v"(a4), "+v"(a5), "+v"(a6), "+v"(a7));
}

// CDNA5 async global -> LDS copy, 128-bit per lane, GVS addressing
// (saddr = uniform row base, vaddr = per-lane byte offset). Tracked by ASYNCcnt.
__device__ __forceinline__ void async_load_b128(unsigned lds_addr, unsigned byte_off,
                                                const float* row_base) {
    asm volatile("global_load_async_to_lds_b128 %0, %1, %2"
                 :: "v"(lds_addr), "v"(byte_off), "s"(row_base)
                 : "memory");
}

__device__ __forceinline__ void wait_asynccnt0() {
    asm volatile("s_wait_asynccnt 0x0" ::: "memory");
}

__global__ __launch_bounds__(TPB)
void iqpool_kernel(const float* __restrict__ x,
                   const float* __restrict__ q_raw,
                   const float* __restrict__ alpha_raw,
                   float* __restrict__ out) {
    // 1028 floats needed (256 windows * stride 4 + K); pad to 1040.
    __shared__ __align__(16) float smem[1040];

    const int      tid   = threadIdx.x;
    const unsigned bid   = blockIdx.x;
    const unsigned row   = bid >> 4;          // b*C + c
    const unsigned chunk = bid & 15u;         // which 256-window slice of the row
    const unsigned c     = row & (C_ - 1);    // C is a power of two

    const float* rowp = x + (size_t)row * L_;

    // Generic LDS pointer low 32 bits == wave LDS byte offset (ISA 10.2).
    const unsigned lds_base  = (unsigned)(unsigned long long)(void*)smem;
    const unsigned base_boff = chunk * (TPB * S_ * 4u);   // chunk * 4096 bytes

    // Stage 1024 floats with 256 async B128 lanes (always in-range)...
    async_load_b128(lds_base + (unsigned)tid * 16u,
                    base_boff + (unsigned)tid * 16u, rowp);
    // ...plus one 16B tail (tile elements 1024..1027), clamped to the row end
    // so the final chunk never reads past the buffer.
    if (tid == 0) {
        unsigned tb = base_boff + 4096u;
        if (tb > ROW_CLAMP) tb = ROW_CLAMP;
        async_load_b128(lds_base + 4096u, tb, rowp);
    }
    wait_asynccnt0();
    __syncthreads();

    // Per-channel parameters (uniform within the block).
    const float qs    = 1.0f / (1.0f + __expf(-q_raw[c]));
    const float alpha = __expf(alpha_raw[c]);
    const float ha    = 0.5f * alpha;                   // tanh argument scale
    const float thr   = (float)K_ * (2.0f * qs - 1.0f); // sum-tanh threshold

    // Window: 8 consecutive floats starting at local element tid*4 (16B aligned).
    const float4 a0 = *(const float4*)&smem[tid * 4];
    const float4 a1 = *(const float4*)&smem[tid * 4 + 4];

    const float xw0 = a0.x, xw1 = a0.y, xw2 = a0.z, xw3 = a0.w;
    const float xw4 = a1.x, xw5 = a1.y, xw6 = a1.z, xw7 = a1.w;

    const float mnv = fminf(fminf(fminf(xw0, xw1), fminf(xw2, xw3)),
                            fminf(fminf(xw4, xw5), fminf(xw6, xw7)));
    const float mxv = fmaxf(fmaxf(fmaxf(xw0, xw1), fmaxf(xw2, xw3)),
                            fmaxf(fmaxf(xw4, xw5), fmaxf(xw6, xw7)));

    // Bisect in u = ha*m space: ha*(2/alpha) == 1, so bounds are +-1 around
    // the scaled extrema, the per-iter argument is u_mid - px_k, and the
    // result is (u_mn + u_mx) * (1/alpha)  [== 0.5*(mn+mx)].
    float umn = ha * mnv - 1.0f;
    float umx = ha * mxv + 1.0f;

    const float px0 = ha * xw0, px1 = ha * xw1, px2 = ha * xw2, px3 = ha * xw3;
    const float px4 = ha * xw4, px5 = ha * xw5, px6 = ha * xw6, px7 = ha * xw7;

    for (int it = 0; it < ITERS_; ++it) {
        const float um = 0.5f * (umn + umx);
        float t0 = um - px0, t1 = um - px1, t2 = um - px2, t3 = um - px3;
        float t4 = um - px4, t5 = um - px5, t6 = um - px6, t7 = um - px7;
        tanh8(t0, t1, t2, t3, t4, t5, t6, t7);
        const float s = ((t0 + t1) + (t2 + t3)) + ((t4 + t5) + (t6 + t7));
        const bool too_high = s > thr;       // mean sigmoid > q
        umn = too_high ? umn : um;
        umx = too_high ? um : umx;
    }

    const unsigned w = chunk * TPB + (unsigned)tid;
    if (w < W_) {
        out[(size_t)row * W_ + w] = (umn + umx) * (1.0f / alpha);
    }
}

extern "C" void kernel_launch(void* const* d_in, const int* in_sizes, int n_in,
                              void* d_out, int out_size, void* d_ws, size_t ws_size,
                              hipStream_t stream) {
    const float* x         = (const float*)d_in[0];
    const float* q_raw     = (const float*)d_in[1];
    const float* alpha_raw = (const float*)d_in[2];
    float* out             = (float*)d_out;

    dim3 grid(B_ * C_ * CHUNKS);   // 32768 blocks: one (row, 256-window chunk) each
    dim3 block(TPB);
    iqpool_kernel<<<grid, block, 0, stream>>>(x, q_raw, alpha_raw, out);
}